// FloodDetectionGraph_29575144800817
// MI455X (gfx1250) — compile-verified
//
#include <hip/hip_runtime.h>
#include <hip/hip_bf16.h>

typedef __attribute__((ext_vector_type(2))) float v2f;
typedef __attribute__((ext_vector_type(8))) float v8f;

#define HID     64
#define NGRAPH  128

// ---------------------------------------------------------------------------
// generic fill
// ---------------------------------------------------------------------------
__global__ void fill_kernel(float* __restrict__ p, long long n, float v) {
  long long i = (long long)blockIdx.x * blockDim.x + threadIdx.x;
  if (i < n) p[i] = v;
}

// ---------------------------------------------------------------------------
// degree count: deg[col[e]] += 1  (deg pre-initialized to 1.0 for self loop)
// ---------------------------------------------------------------------------
__global__ void degree_kernel(const int* __restrict__ cols,
                              float* __restrict__ deg, int E) {
  int e = blockIdx.x * blockDim.x + threadIdx.x;
  if (e < E) atomicAdd(&deg[cols[e]], 1.0f);
}

__global__ void dinv_kernel(const float* __restrict__ deg,
                            float* __restrict__ dinv, int N) {
  int i = blockIdx.x * blockDim.x + threadIdx.x;
  if (i < N) {
    float d = deg[i];
    dinv[i] = (d > 0.0f) ? rsqrtf(d) : 0.0f;
  }
}

// ---------------------------------------------------------------------------
// h = x @ W via V_WMMA_F32_16X16X4_F32 (wave32, one wave per 16-row tile)
//
// A 16x4 f32 layout : lanes 0-15 -> M=lane, K={0,1} in v0,v1
//                     lanes 16-31 -> M=lane-16, K={2,3} in v0,v1
// B 4x16 f32 layout : lanes 0-15 -> N=lane, K={0,1} in v0,v1
//                     lanes 16-31 -> N=lane-16, K={2,3} in v0,v1
// C/D 16x16 f32     : VGPR r, lane L -> (M = r + (L>=16?8:0), N = L&15)
//
// W is staged in LDS pair-interleaved: Wp[(k>>1)*64 + n] = {W[k][n], W[k+1][n]}
// so each B fragment is one aligned ds_load_b64 (no VALU repacking).
// ---------------------------------------------------------------------------
__global__ __launch_bounds__(256) void gemm_xw_wmma(
    const float* __restrict__ x,      // [N,64]
    const float* __restrict__ W,      // [64,64]
    float* __restrict__ h,            // [N,64]
    int ntiles)                       // N/16
{
  __shared__ v2f Wp[32 * HID];        // 16 KB of the 320 KB WGP LDS
  for (int idx = threadIdx.x; idx < HID * HID; idx += blockDim.x) {
    const int k = idx >> 6;
    const int n = idx & 63;
    ((float*)Wp)[((((k >> 1) << 6) + n) << 1) + (k & 1)] = W[idx];
  }
  __syncthreads();

  const int wave = threadIdx.x >> 5;            // wave32
  const int lane = threadIdx.x & 31;
  const int tile = blockIdx.x * 8 + wave;
  if (tile >= ntiles) return;                   // wave-uniform -> EXEC all 1s inside

  const int l16   = lane & 15;
  const int khalf = lane >> 4;                  // 0 or 1

  v8f acc[4] = {};                              // 4 n-tiles of 16 columns

  const v2f* xrow = (const v2f*)(x + (size_t)(tile * 16 + l16) * HID);

  #pragma unroll
  for (int kc = 0; kc < 16; ++kc) {             // K = 64 in chunks of 4
    const int kpair = kc * 2 + khalf;           // index of (k0, k0+1) pair
    const v2f a = xrow[kpair];                  // global_load_b64
    #pragma unroll
    for (int nt = 0; nt < 4; ++nt) {
      const v2f bm = Wp[(kpair << 6) + nt * 16 + l16];   // ds_load_b64
      acc[nt] = __builtin_amdgcn_wmma_f32_16x16x4_f32(
          /*neg_a=*/false, a, /*neg_b=*/false, bm,
          /*c_mod=*/(short)0, acc[nt], /*reuse_a=*/false, /*reuse_b=*/false);
    }
  }

  const int mrow = tile * 16 + khalf * 8;
  #pragma unroll
  for (int nt = 0; nt < 4; ++nt) {
    #pragma unroll
    for (int r = 0; r < 8; ++r) {
      h[(size_t)(mrow + r) * HID + nt * 16 + l16] = acc[nt][r];
    }
  }
}

// ---------------------------------------------------------------------------
// edge scatter: agg[col] += dinv[row]*dinv[col] * h[row]
// 16 threads per edge, float4 gather + 4 f32 atomics each.
// Prefetch the h row of an edge ~128 ahead to hide the idx->gather chain.
// ---------------------------------------------------------------------------
__global__ void scatter_kernel(const int* __restrict__ rows,
                               const int* __restrict__ cols,
                               const float* __restrict__ dinv,
                               const float* __restrict__ h,
                               float* __restrict__ agg, int E) {
  long long gid = (long long)blockIdx.x * blockDim.x + threadIdx.x;
  long long e = gid >> 4;
  int q = (int)(gid & 15);
  if (e >= E) return;

  long long e2 = e + 128;
  if (q == 0 && e2 < E) {
    __builtin_prefetch(h + (size_t)rows[e2] * HID, 0, 0);  // global_prefetch_b8
  }

  int r = rows[e];
  int c = cols[e];
  float nrm = dinv[r] * dinv[c];
  float4 v = *((const float4*)(h + (size_t)r * HID) + q);
  float* o = agg + (size_t)c * HID + q * 4;
  atomicAdd(o + 0, nrm * v.x);
  atomicAdd(o + 1, nrm * v.y);
  atomicAdd(o + 2, nrm * v.z);
  atomicAdd(o + 3, nrm * v.w);
}

// ---------------------------------------------------------------------------
// per node: val = agg + dinv^2 * h (self loop) + b; relu; pool into gsum/gcnt
// ---------------------------------------------------------------------------
__global__ void finalize_kernel(const float* __restrict__ agg,
                                const float* __restrict__ h,
                                const float* __restrict__ dinv,
                                const float* __restrict__ b,
                                const int* __restrict__ batch,
                                float* __restrict__ gsum,
                                float* __restrict__ gcnt, int N) {
  long long gid = (long long)blockIdx.x * blockDim.x + threadIdx.x;
  if (gid >= (long long)N * HID) return;
  int i = (int)(gid >> 6);
  int j = (int)(gid & 63);
  float di = dinv[i];
  float val = agg[gid] + di * di * h[gid] + b[j];
  val = fmaxf(val, 0.0f);
  int g = batch[i];
  atomicAdd(&gsum[(size_t)g * HID + j], val);
  if (j == 0) atomicAdd(&gcnt[g], 1.0f);
}

// ---------------------------------------------------------------------------
// out[g,o] = (sum_j gsum[g,j]*lin_W[j,o]) / max(cnt,1) + lin_b[o]
// ---------------------------------------------------------------------------
__global__ void head_kernel(const float* __restrict__ gsum,
                            const float* __restrict__ gcnt,
                            const float* __restrict__ lin_W,   // [64,2]
                            const float* __restrict__ lin_b,   // [2]
                            float* __restrict__ out) {         // [128,2]
  int t = threadIdx.x;
  if (t >= NGRAPH * 2) return;
  int g = t >> 1;
  int o = t & 1;
  float cnt = fmaxf(gcnt[g], 1.0f);
  float s = 0.0f;
  #pragma unroll
  for (int j = 0; j < HID; ++j) s += gsum[(size_t)g * HID + j] * lin_W[j * 2 + o];
  out[t] = s / cnt + lin_b[o];
}

// ---------------------------------------------------------------------------
extern "C" void kernel_launch(void* const* d_in, const int* in_sizes, int n_in,
                              void* d_out, int out_size, void* d_ws, size_t ws_size,
                              hipStream_t stream) {
  const float* x     = (const float*)d_in[0];   // [N,64]
  const int*   ei    = (const int*)d_in[1];     // [2,E] flattened
  const int*   batch = (const int*)d_in[2];     // [N]
  const float* W     = (const float*)d_in[3];   // [64,64]
  const float* b     = (const float*)d_in[4];   // [64]
  const float* lin_W = (const float*)d_in[5];   // [64,2]
  const float* lin_b = (const float*)d_in[6];   // [2]
  float* out = (float*)d_out;

  const int N = in_sizes[0] / HID;              // 50000
  const int E = in_sizes[1] / 2;                // 800000
  const int* rows = ei;                         // edge_index[0] = sources
  const int* cols = ei + E;                     // edge_index[1] = targets

  // workspace layout (floats)
  float* ws   = (float*)d_ws;
  float* deg  = ws;                              // [N]
  float* dinv = deg + N;                         // [N]
  float* h    = dinv + N;                        // [N*64]
  float* agg  = h + (size_t)N * HID;             // [N*64]
  float* gsum = agg + (size_t)N * HID;           // [128*64]
  float* gcnt = gsum + NGRAPH * HID;             // [128]

  const int T = 256;

  // init scratch (every launch — harness poisons once, never restores)
  fill_kernel<<<(N + T - 1) / T, T, 0, stream>>>(deg, N, 1.0f);  // self loop
  {
    long long n = (long long)N * HID;
    fill_kernel<<<(unsigned)((n + T - 1) / T), T, 0, stream>>>(agg, n, 0.0f);
  }
  fill_kernel<<<(NGRAPH * HID + NGRAPH + T - 1) / T, T, 0, stream>>>(
      gsum, NGRAPH * HID + NGRAPH, 0.0f);        // gsum + gcnt contiguous

  degree_kernel<<<(E + T - 1) / T, T, 0, stream>>>(cols, deg, E);
  dinv_kernel<<<(N + T - 1) / T, T, 0, stream>>>(deg, dinv, N);

  const int ntiles = N / 16;                     // 3125 (N divisible by 16)
  gemm_xw_wmma<<<(ntiles + 7) / 8, 256, 0, stream>>>(x, W, h, ntiles);

  {
    long long work = (long long)E * 16;
    scatter_kernel<<<(unsigned)((work + T - 1) / T), T, 0, stream>>>(
        rows, cols, dinv, h, agg, E);
  }

  {
    long long n = (long long)N * HID;
    finalize_kernel<<<(unsigned)((n + T - 1) / T), T, 0, stream>>>(
        agg, h, dinv, b, batch, gsum, gcnt, N);
  }

  head_kernel<<<1, 256, 0, stream>>>(gsum, gcnt, lin_W, lin_b, out);
}